// ScientificModel_5961414607421
// MI455X (gfx1250) — compile-verified
//
#include <hip/hip_runtime.h>

typedef __attribute__((ext_vector_type(2))) float v2f;
typedef __attribute__((ext_vector_type(8))) float v8f;

#define HDIM 4096
#define WDIM 4096
#define NPIX (HDIM * WDIM)                  // 16,777,216
#define NTILES ((HDIM / 16) * (WDIM / 16))  // 65536 tiles of 16x16, one per wave32
#define NBUCKET 64                          // contention-spreading buckets for reductions

// Compute a 16x16 tile of conv(p, 3x3 filter, pad=1) with V_WMMA_F32_16X16X4_F32.
//   out[m][n] = sum_dh sum_c A_dh[m][c] * B_dh[c][n]
// A_dh: 16x20 image slab (cols wb-1 .. wb+18, zero padded) at rows hb+m+dh-1
// B_dh[c][n] = f[dh][c-n] for c-n in {0,1,2}, else 0 (banded shift matrix).
// f32 WMMA operand layout: (vgpr v, lane = x + 16*g) holds K = v + 2*g.
// C/D layout: vgpr r, lane group g -> element (M = r + 8*g, N = x).
__device__ __forceinline__ v8f conv_tile(const float* __restrict__ p,
                                         const float* __restrict__ filt,
                                         int hb, int wb, int x, int g) {
    // ---- B operands: band compares hoisted, filter indexed statically ----
    v2f B[15];
#pragma unroll
    for (int kc = 0; kc < 5; ++kc) {
#pragma unroll
        for (int v = 0; v < 2; ++v) {
            const int fi = kc * 4 + v + 2 * g - x;   // per-lane band position
            const bool m0 = (fi == 0), m1 = (fi == 1), m2 = (fi == 2);
#pragma unroll
            for (int dh = 0; dh < 3; ++dh) {
                float b = 0.0f;
                b = m0 ? filt[dh * 3 + 0] : b;       // static idx -> sgpr operand
                b = m1 ? filt[dh * 3 + 1] : b;
                b = m2 ? filt[dh * 3 + 2] : b;
                B[dh * 5 + kc][v] = b;
            }
        }
    }

    // ---- A operands: all loads issued up front (compiler merges to b64 clauses) ----
    v2f A[15];
    // wave-uniform branch (hb, wb uniform per wave): no exec divergence
    const bool interior = (hb >= 1) && (hb <= HDIM - 17) && (wb >= 1) && (wb <= WDIM - 19);
    if (interior) {
#pragma unroll
        for (int dh = 0; dh < 3; ++dh) {
            const float* rowp = p + (((hb + x + dh - 1) << 12) + wb + 2 * g - 1);
#pragma unroll
            for (int kc = 0; kc < 5; ++kc) {
                A[dh * 5 + kc][0] = rowp[kc * 4];      // immediate-offset loads
                A[dh * 5 + kc][1] = rowp[kc * 4 + 1];
            }
        }
    } else {
#pragma unroll
        for (int dh = 0; dh < 3; ++dh) {
            const int row = hb + x + dh - 1;
            const bool rok = (unsigned)row < (unsigned)HDIM;
#pragma unroll
            for (int kc = 0; kc < 5; ++kc) {
#pragma unroll
                for (int v = 0; v < 2; ++v) {
                    const int col = wb + kc * 4 + 2 * g - 1 + v;
                    A[dh * 5 + kc][v] =
                        (rok && (unsigned)col < (unsigned)WDIM) ? p[(row << 12) + col] : 0.0f;
                }
            }
        }
    }

    // ---- 15 WMMAs in 3 independent accumulator chains (one per filter row) ----
    v8f C0 = {}, C1 = {}, C2 = {};
#pragma unroll
    for (int kc = 0; kc < 5; ++kc) {
        C0 = __builtin_amdgcn_wmma_f32_16x16x4_f32(false, A[kc], false, B[kc],
                                                   (short)0, C0, false, false);
        C1 = __builtin_amdgcn_wmma_f32_16x16x4_f32(false, A[5 + kc], false, B[5 + kc],
                                                   (short)0, C1, false, false);
        C2 = __builtin_amdgcn_wmma_f32_16x16x4_f32(false, A[10 + kc], false, B[10 + kc],
                                                   (short)0, C2, false, false);
    }
    v8f C;
#pragma unroll
    for (int r = 0; r < 8; ++r) C[r] = C0[r] + (C1[r] + C2[r]);
    return C;
}

__global__ void cg_init(float* __restrict__ buckets) {
    if (threadIdx.x < NBUCKET * 4) buckets[threadIdx.x] = 0.0f;
}

// Pass 1: conv via WMMA + four global dot products into 64x4 buckets:
//   { <r0,r0>, <p,conv>, <r0,conv>, <conv,conv> }
__global__ void __launch_bounds__(256) cg_pass1(const float* __restrict__ r0,
                                                const float* __restrict__ p,
                                                const float* __restrict__ filt,
                                                float* __restrict__ buckets) {
    const int lane = threadIdx.x & 31;
    const int wid  = threadIdx.x >> 5;
    const int tile = blockIdx.x * 8 + wid;
    const int hb = (tile >> 8) << 4;
    const int wb = (tile & 255) << 4;
    const int x = lane & 15;
    const int g = lane >> 4;

    const v8f C = conv_tile(p, filt, hb, wb, x, g);

    float rr = 0.f, pa = 0.f, rc = 0.f, cc = 0.f;
#pragma unroll
    for (int r = 0; r < 8; ++r) {
        const int idx = ((hb + r + 8 * g) << 12) + (wb + x);
        const float cv = C[r];
        const float rv = r0[idx];
        const float pv = p[idx];
        rr = fmaf(rv, rv, rr);
        pa = fmaf(pv, cv, pa);
        rc = fmaf(rv, cv, rc);
        cc = fmaf(cv, cv, cc);
    }
    // wave32 butterfly reduction
#pragma unroll
    for (int m = 16; m >= 1; m >>= 1) {
        rr += __shfl_xor(rr, m, 32);
        pa += __shfl_xor(pa, m, 32);
        rc += __shfl_xor(rc, m, 32);
        cc += __shfl_xor(cc, m, 32);
    }
    __shared__ float sred[8][4];
    if (lane == 0) {
        sred[wid][0] = rr; sred[wid][1] = pa; sred[wid][2] = rc; sred[wid][3] = cc;
    }
    __syncthreads();
    if (threadIdx.x < 4) {
        float t = 0.f;
#pragma unroll
        for (int w = 0; w < 8; ++w) t += sred[w][threadIdx.x];
        // spread contention over 64 bucket sets: ~128 serialized RMWs per address
        atomicAdd(&buckets[(blockIdx.x & (NBUCKET - 1)) * 4 + threadIdx.x], t);
    }
}

// Pass 2: reduce buckets in-block, alpha/beta closed form, conv (WMMA), fused update.
//   r1      = r0 - alpha*conv
//   p_new   = r1 + beta*p
//   phi_new = phi + alpha*p
//   r1_sum  = rr0 - 2a<r0,conv> + a^2<conv,conv>
__global__ void __launch_bounds__(256) cg_pass2(const float* __restrict__ r0,
                                                const float* __restrict__ p,
                                                const float* __restrict__ phi,
                                                const float* __restrict__ filt,
                                                const float* __restrict__ buckets,
                                                float* __restrict__ out) {
    const int lane = threadIdx.x & 31;
    const int wid  = threadIdx.x >> 5;
    const int tile = blockIdx.x * 8 + wid;
    const int hb = (tile >> 8) << 4;
    const int wb = (tile & 255) << 4;
    const int x = lane & 15;
    const int g = lane >> 4;

    // Start the conv loads first so they are in flight across the barrier below.
    const v8f C = conv_tile(p, filt, hb, wb, x, g);

    __shared__ float sacc[4];
    if (threadIdx.x < 4) {
        float t = 0.f;
#pragma unroll
        for (int b = 0; b < NBUCKET; ++b) t += buckets[b * 4 + threadIdx.x];
        sacc[threadIdx.x] = t;
    }
    __syncthreads();
    const float rr0 = sacc[0], pAp = sacc[1], rAc = sacc[2], ccv = sacc[3];
    const float alpha = rr0 / pAp;
    const float r1sum = fmaf(alpha * alpha, ccv, fmaf(-2.0f * alpha, rAc, rr0));
    const float beta  = r1sum / rr0;

    float* __restrict__ out_r1  = out;
    float* __restrict__ out_p   = out + NPIX;
    float* __restrict__ out_phi = out + 2 * (size_t)NPIX;
#pragma unroll
    for (int r = 0; r < 8; ++r) {
        const int idx = ((hb + r + 8 * g) << 12) + (wb + x);
        const float rv = r0[idx];
        const float pv = p[idx];
        const float ph = __builtin_nontemporal_load(&phi[idx]);   // read exactly once
        const float r1 = fmaf(-alpha, C[r], rv);
        __builtin_nontemporal_store(r1, &out_r1[idx]);            // outputs never re-read
        __builtin_nontemporal_store(fmaf(beta, pv, r1), &out_p[idx]);
        __builtin_nontemporal_store(fmaf(alpha, pv, ph), &out_phi[idx]);
    }
    if (blockIdx.x == 0 && threadIdx.x == 0)
        out[(size_t)3 * NPIX] = r1sum;
}

extern "C" void kernel_launch(void* const* d_in, const int* in_sizes, int n_in,
                              void* d_out, int out_size, void* d_ws, size_t ws_size,
                              hipStream_t stream) {
    const float* r0  = (const float*)d_in[0];
    const float* p   = (const float*)d_in[1];
    const float* phi = (const float*)d_in[2];
    const float* flt = (const float*)d_in[3];
    float* out = (float*)d_out;
    float* buckets = (float*)d_ws;   // NBUCKET x 4 partial sums

    const int blocks = NTILES / 8;   // 8 wave32 per 256-thread block, 1 tile per wave

    cg_init<<<1, 256, 0, stream>>>(buckets);
    cg_pass1<<<blocks, 256, 0, stream>>>(r0, p, flt, buckets);
    cg_pass2<<<blocks, 256, 0, stream>>>(r0, p, phi, flt, buckets, out);
}